// GlobalAttentionPooling_52329881534841
// MI455X (gfx1250) — compile-verified
//
#include <hip/hip_runtime.h>
#include <hip/hip_bf16.h>
#include <math.h>

typedef __attribute__((ext_vector_type(2))) float v2f;
typedef __attribute__((ext_vector_type(8))) float v8f;

#define DIM   256   // D == H == 256
#define NSEG  256   // B
#define TILE  32    // nodes per LDS tile

// ---------------------------------------------------------------------------
// Pass A: per-segment online-softmax gate + weighted feature pooling.
// One block per segment. Thread t owns feature channel t of the pooled vector.
// Writes P[b][d] = sum_n softmax(gate)_n * x[n][d], and flag[b] = (count>0).
// ---------------------------------------------------------------------------
__global__ __launch_bounds__(256) void gap_pool_kernel(
    const float* __restrict__ x,    // [N, 256]
    const float* __restrict__ Wg,   // [256]
    const float* __restrict__ bg,   // [1]
    const int*   __restrict__ seg,  // [N] sorted
    int N,
    float* __restrict__ P,          // [256, 256]
    float* __restrict__ flag)       // [256]
{
    __shared__ float lds_x[TILE * DIM];   // 32 KB tile
    __shared__ float lds_wg[DIM];
    __shared__ float lds_part[256];
    __shared__ float lds_gate[TILE];
    __shared__ float lds_e[TILE];
    __shared__ float sm_m, sm_s, sm_c;
    __shared__ int   sm_range[2];

    const int t = threadIdx.x;
    const int b = blockIdx.x;

    lds_wg[t] = Wg[t];
    const float bgv = bg[0];

    if (t < 2) {
        // lower_bound(seg, b + t)
        int key = b + t;
        int lo = 0, hi = N;
        while (lo < hi) {
            int mid = (lo + hi) >> 1;
            if (seg[mid] < key) lo = mid + 1; else hi = mid;
        }
        sm_range[t] = lo;
    }
    if (t == 0) { sm_m = -INFINITY; sm_s = 0.0f; sm_c = 0.0f; }
    __syncthreads();

    const int start = sm_range[0];
    const int end   = sm_range[1];

    float v = 0.0f;   // pooled accumulator for channel t

    for (int base = start; base < end; base += TILE) {
        const int cnt = min(TILE, end - base);

        // ---- cooperative tile load: 32 nodes x 256 floats = 2048 float4 ----
        const float4* src = (const float4*)(x + (size_t)base * DIM);
        #pragma unroll
        for (int i = 0; i < 8; ++i) {
            int j = t + i * 256;          // float4 index in tile
            int n = j >> 6;               // 64 float4 per node
            float4 val = make_float4(0.f, 0.f, 0.f, 0.f);
            if (n < cnt) val = src[j];
            ((float4*)lds_x)[j] = val;
        }
        // prefetch next tile (gfx1250 global_prefetch_b8)
        if (base + TILE < end) {
            __builtin_prefetch(x + (size_t)(base + TILE) * DIM + (t << 5), 0, 1);
        }
        __syncthreads();

        // ---- gate partial dots: node n = t/8, slice = t%8 (32 MACs each) ----
        {
            int n = t >> 3, sl = t & 7;
            const float* xr = lds_x + n * DIM + sl * 32;
            const float* wr = lds_wg + sl * 32;
            float acc = 0.0f;
            #pragma unroll
            for (int d = 0; d < 32; ++d) acc = fmaf(xr[d], wr[d], acc);
            lds_part[t] = acc;
        }
        __syncthreads();

        if (t < TILE) {
            float g = bgv;
            #pragma unroll
            for (int sl = 0; sl < 8; ++sl) g += lds_part[t * 8 + sl];
            lds_gate[t] = (t < cnt) ? g : -INFINITY;
        }
        __syncthreads();

        if (t == 0) {
            float tm = sm_m;
            for (int n = 0; n < TILE; ++n) tm = fmaxf(tm, lds_gate[n]);
            sm_c = __expf(sm_m - tm);   // exp(-inf)=0 on first tile
            sm_m = tm;
        }
        __syncthreads();

        if (t < TILE) lds_e[t] = __expf(lds_gate[t] - sm_m);  // -inf -> 0 pad
        __syncthreads();

        const float c = sm_c;
        float acc = 0.0f;
        #pragma unroll
        for (int n = 0; n < TILE; ++n)
            acc = fmaf(lds_e[n], lds_x[n * DIM + t], acc);
        v = v * c + acc;

        if (t == 0) {
            float es = 0.0f;
            for (int n = 0; n < TILE; ++n) es += lds_e[n];
            sm_s = sm_s * c + es;
        }
        __syncthreads();   // protect lds_x / sm_* before next tile
    }

    const float s = sm_s;
    P[(size_t)b * DIM + t] = (end > start) ? (v / s) : 0.0f;
    if (t == 0) flag[b] = (end > start) ? 1.0f : 0.0f;
}

// ---------------------------------------------------------------------------
// Pass B: Out[type] = P[type] @ Wf[type]^T + bf[type] * flag   (256x256x256)
// One wave per 16x16 C tile, fp32 WMMA 16x16x4, K-loop of 64 steps.
// A-lane layout: lane<16 -> row=lane, K-pair {k,k+1}; lane>=16 -> K-pair {k+2,k+3}
// B (=Wf^T) same pairing over Wf rows (columns of B).
// C VGPR j: lanes0-15 -> row j, lanes16-31 -> row j+8.
// ---------------------------------------------------------------------------
__global__ __launch_bounds__(256) void gap_gemm_kernel(
    const float* __restrict__ Pbase,   // [3][256][256] in workspace
    const float* __restrict__ flags,   // [3][256]
    const float* __restrict__ Wf0, const float* __restrict__ Wf1, const float* __restrict__ Wf2,
    const float* __restrict__ bf0, const float* __restrict__ bf1, const float* __restrict__ bf2,
    float* __restrict__ out)           // [3][256][256]
{
    const int type = blockIdx.y;
    const float* Wf = (type == 0) ? Wf0 : ((type == 1) ? Wf1 : Wf2);
    const float* bf = (type == 0) ? bf0 : ((type == 1) ? bf1 : bf2);
    const float* Pt = Pbase + (size_t)type * DIM * NSEG;
    const float* fl = flags + (size_t)type * NSEG;
    float*       Ot = out   + (size_t)type * DIM * NSEG;

    const int wave = threadIdx.x >> 5;            // 0..7
    const int lane = threadIdx.x & 31;
    const int tile = blockIdx.x * 8 + wave;       // 0..255
    const int tm = (tile >> 4) * 16;              // C row block
    const int tn = (tile & 15) * 16;              // C col block

    const int idx = lane & 15;
    const int kp  = (lane >> 4) << 1;             // 0 or 2

    const float* arow = Pt + (size_t)(tm + idx) * DIM;   // P row  (A)
    const float* brow = Wf + (size_t)(tn + idx) * DIM;   // Wf row (B column)

    v8f c = {};
    #pragma unroll 8
    for (int k = 0; k < DIM; k += 4) {
        v2f a = *(const v2f*)(arow + k + kp);
        v2f bb = *(const v2f*)(brow + k + kp);
        c = __builtin_amdgcn_wmma_f32_16x16x4_f32(
                /*neg_a=*/false, a, /*neg_b=*/false, bb,
                /*c_mod=*/(short)0, c, /*reuse_a=*/false, /*reuse_b=*/false);
    }

    const int rbase = tm + ((lane >> 4) << 3);    // rows j / j+8 split
    const int col   = tn + idx;
    const float bias = bf[col];
    #pragma unroll
    for (int j = 0; j < 8; ++j) {
        float f = fl[rbase + j];                  // suppress bias on empty segments
        Ot[(size_t)(rbase + j) * DIM + col] = c[j] + bias * f;
    }
}

// ---------------------------------------------------------------------------
extern "C" void kernel_launch(void* const* d_in, const int* in_sizes, int n_in,
                              void* d_out, int out_size, void* d_ws, size_t ws_size,
                              hipStream_t stream) {
    (void)n_in; (void)out_size; (void)ws_size;
    // setup_inputs dict order: per type {feat, Wf, bf, Wg, bg, seg}, then num_segments
    const float* feat[3] = { (const float*)d_in[0],  (const float*)d_in[6],  (const float*)d_in[12] };
    const float* Wf[3]   = { (const float*)d_in[1],  (const float*)d_in[7],  (const float*)d_in[13] };
    const float* bfp[3]  = { (const float*)d_in[2],  (const float*)d_in[8],  (const float*)d_in[14] };
    const float* Wg[3]   = { (const float*)d_in[3],  (const float*)d_in[9],  (const float*)d_in[15] };
    const float* bg[3]   = { (const float*)d_in[4],  (const float*)d_in[10], (const float*)d_in[16] };
    const int*   seg[3]  = { (const int*)d_in[5],    (const int*)d_in[11],   (const int*)d_in[17] };
    const int    N[3]    = { in_sizes[5], in_sizes[11], in_sizes[17] };

    float* Pws   = (float*)d_ws;                        // 3 * 256 * 256 floats
    float* flags = Pws + 3 * DIM * NSEG;                // 3 * 256 floats

    for (int t = 0; t < 3; ++t) {
        gap_pool_kernel<<<NSEG, 256, 0, stream>>>(
            feat[t], Wg[t], bg[t], seg[t], N[t],
            Pws + (size_t)t * DIM * NSEG, flags + (size_t)t * NSEG);
    }
    gap_gemm_kernel<<<dim3(32, 3), 256, 0, stream>>>(
        Pws, flags, Wf[0], Wf[1], Wf[2], bfp[0], bfp[1], bfp[2], (float*)d_out);
}